// RandomResizedCropLayer_8452495639045
// MI455X (gfx1250) — compile-verified
//
#include <hip/hip_runtime.h>

// RandomResizedCrop (bilinear, reflect padding) for N=128, C=3, H=W=256, f32.
// Memory-bound: target the 23.3 TB/s HBM roofline.
//   - global_load_async_to_lds_b32 stages the two source rows per channel as
//     fully coalesced 1KB row reads (ASYNCcnt-tracked, no VGPR round-trip)
//   - divergent bilinear taps become LDS gathers (compiler pairs them into
//     ds_load_2addr_stride64 ops since the two rows are 1024B apart)
//   - output is write-once streaming -> non-temporal stores
//   - reflect math uses exact power-of-two floored-mod (t - floor(t/512)*512),
//     which is bit-identical to jnp's float `%` and avoids the branchy OCML
//     fmodf libcall entirely -> straight-line code.

#define IMG 256
#define CH 3
#define PLANE (IMG * IMG)          // 65536 floats per channel plane

__device__ __forceinline__ float reflect_clip256(float v) {
    const float span = 256.0f;     // 2*span = 512 = 2^9 -> exact fp32 mod
    float t = fabsf(v + 0.5f);
    t = t - floorf(t * (1.0f / 512.0f)) * 512.0f;   // floored mod, exact
    t = (t > span) ? (512.0f - t) : t;
    t -= 0.5f;
    return fminf(fmaxf(t, 0.0f), span - 1.0f);
}

__global__ __launch_bounds__(256)
void rrc_kernel(const float* __restrict__ in,
                const float* __restrict__ whbias,
                float* __restrict__ out) {
    // Block = one output row (n, y). Thread = one x.
    __shared__ float lds[CH * 2][IMG];    // [c*2 + row01][x], 6 KB

    const int x = threadIdx.x;
    const int y = blockIdx.x & (IMG - 1);
    const int n = blockIdx.x >> 8;

    const float w_s = whbias[n * 4 + 0];
    const float h_s = whbias[n * 4 + 1];
    const float w_b = whbias[n * 4 + 2];
    const float h_b = whbias[n * 4 + 3];

    // ---- y mapping (block-uniform -> scalar-float pipe) ----
    const float ys = ((float)y + 0.5f) * (2.0f / (float)IMG) - 1.0f;
    const float gy = h_s * ys + h_b;
    float fy = ((gy + 1.0f) * (float)IMG - 1.0f) * 0.5f;
    fy = reflect_clip256(fy);
    const float fy0 = floorf(fy);
    const float wy  = fy - fy0;
    int iy0 = (int)fy0;
    iy0 = min(max(iy0, 0), IMG - 1);
    const int iy1 = min(iy0 + 1, IMG - 1);

    // ---- async-stage the two source rows for all 3 channels into LDS ----
    const float* base_n = in + (size_t)n * CH * PLANE;
    const unsigned lds_base = (unsigned)(unsigned long long)(&lds[0][0]);
#pragma unroll
    for (int c = 0; c < CH; ++c) {
        const float* g0 = base_n + c * PLANE + iy0 * IMG + x;
        const float* g1 = base_n + c * PLANE + iy1 * IMG + x;
        const unsigned l0 = lds_base + (unsigned)((c * 2 + 0) * IMG * 4 + x * 4);
        const unsigned l1 = lds_base + (unsigned)((c * 2 + 1) * IMG * 4 + x * 4);
        asm volatile("global_load_async_to_lds_b32 %0, %1, off"
                     :: "v"(l0), "v"(g0) : "memory");
        asm volatile("global_load_async_to_lds_b32 %0, %1, off"
                     :: "v"(l1), "v"(g1) : "memory");
    }

    // ---- x mapping (VALU; overlaps the in-flight async transfers) ----
    const float xs = ((float)x + 0.5f) * (2.0f / (float)IMG) - 1.0f;
    const float gx = w_s * xs + w_b;
    float fx = ((gx + 1.0f) * (float)IMG - 1.0f) * 0.5f;
    fx = reflect_clip256(fx);
    const float fx0 = floorf(fx);
    const float wx  = fx - fx0;
    int ix0 = (int)fx0;
    ix0 = min(max(ix0, 0), IMG - 1);
    const int ix1 = min(ix0 + 1, IMG - 1);

    const float w00 = (1.0f - wx) * (1.0f - wy);
    const float w01 = wx * (1.0f - wy);
    const float w10 = (1.0f - wx) * wy;
    const float w11 = wx * wy;

    // Drain this wave's async transfers, then barrier so all waves' staged
    // rows are visible in LDS.
    asm volatile("s_wait_asynccnt 0" ::: "memory");
    __syncthreads();

    // ---- gather taps from LDS, blend, stream out (non-temporal) ----
    const size_t obase = (size_t)n * CH * PLANE + (size_t)y * IMG + x;
#pragma unroll
    for (int c = 0; c < CH; ++c) {
        const float v00 = lds[c * 2 + 0][ix0];
        const float v01 = lds[c * 2 + 0][ix1];
        const float v10 = lds[c * 2 + 1][ix0];
        const float v11 = lds[c * 2 + 1][ix1];
        const float r = v00 * w00 + v01 * w01 + v10 * w10 + v11 * w11;
        __builtin_nontemporal_store(r, out + obase + (size_t)c * PLANE);
    }
}

extern "C" void kernel_launch(void* const* d_in, const int* in_sizes, int n_in,
                              void* d_out, int out_size, void* d_ws, size_t ws_size,
                              hipStream_t stream) {
    const float* in     = (const float*)d_in[0];   // (128, 3, 256, 256) f32
    const float* whbias = (const float*)d_in[1];   // (128, 4) f32
    float* out = (float*)d_out;                    // (128, 3, 256, 256) f32

    const int N = 128;
    dim3 grid(N * IMG);   // one block per (n, y) output row
    dim3 block(256);
    rrc_kernel<<<grid, block, 0, stream>>>(in, whbias, out);
}